// GraphSage_71854802862598
// MI455X (gfx1250) — compile-verified
//
#include <hip/hip_runtime.h>

typedef __attribute__((ext_vector_type(2))) float v2f;
typedef __attribute__((ext_vector_type(4))) float v4f;
typedef __attribute__((ext_vector_type(8))) float v8f;

#define D 128
#define LDW 144   // padded LDS row stride (floats): 144 % 64 == 16 banks/row shift

// ---------------- zeroing ----------------
__global__ void zero_v4(float* __restrict__ p, long long n4) {
    long long i = (long long)blockIdx.x * blockDim.x + threadIdx.x;
    long long stride = (long long)gridDim.x * blockDim.x;
    for (; i < n4; i += stride) {
        *(v4f*)(p + 4 * i) = (v4f){0.f, 0.f, 0.f, 0.f};
    }
}

// ---------------- degree ----------------
__global__ void degree_kernel(const int* __restrict__ row, int* __restrict__ deg, int nE) {
    int i = blockIdx.x * blockDim.x + threadIdx.x;
    int stride = gridDim.x * blockDim.x;
    for (; i < nE; i += stride) atomicAdd(&deg[row[i]], 1);
}

__global__ void invdeg_kernel(const int* __restrict__ deg, float* __restrict__ inv, int n) {
    int i = blockIdx.x * blockDim.x + threadIdx.x;
    if (i < n) {
        int d = deg[i];
        inv[i] = (d > 0) ? (1.0f / (float)d) : 0.0f;
    }
}

// ---------------- edge aggregation: agg[row] += feat[col] ----------------
// one wave per edge; each lane moves 4 floats (128 / 32 lanes)
__global__ void agg_kernel(const float* __restrict__ feat, const int* __restrict__ row,
                           const int* __restrict__ col, float* __restrict__ agg, int nE) {
    int wave   = (blockIdx.x * blockDim.x + threadIdx.x) >> 5;
    int lane   = threadIdx.x & 31;
    int nWaves = (gridDim.x * blockDim.x) >> 5;
    for (int e = wave; e < nE; e += nWaves) {
        int r = row[e];
        int c = col[e];
        v4f v = *(const v4f*)(feat + (long long)c * D + lane * 4);
        float* dst = agg + (long long)r * D + lane * 4;
        unsafeAtomicAdd(dst + 0, v.x);
        unsafeAtomicAdd(dst + 1, v.y);
        unsafeAtomicAdd(dst + 2, v.z);
        unsafeAtomicAdd(dst + 3, v.w);
    }
}

// ---------------- fused SAGE layer GEMM: OUT = act([X | AGG*invdeg] @ W + b) ----------------
// W is 256 x 128 row-major. Block = 8 waves; each wave owns a 16-row strip,
// all 8 column tiles (16x16 each) via V_WMMA_F32_16X16X4_F32.
__global__ __launch_bounds__(256) void sage_gemm(
    const float* __restrict__ X, const float* __restrict__ AGG,
    const float* __restrict__ invdeg, const float* __restrict__ W,
    const float* __restrict__ bias, float* __restrict__ OUT,
    int n, int relu) {
    __shared__ float sW[64 * LDW];   // 36 KB chunk of W

    const int t    = threadIdx.x;
    const int lane = t & 31;
    const int wave = t >> 5;
    const int half = lane >> 4;   // 0: lanes 0-15, 1: lanes 16-31
    const int nl   = lane & 15;
    const int m0   = blockIdx.x * 128 + wave * 16;
    const bool active = (m0 < n);   // uniform per wave -> EXEC all-1s inside branch

    float invd = 0.0f;
    if (active) invd = invdeg[m0 + nl];

    v8f acc[8] = {};

    for (int kc = 0; kc < 256; kc += 64) {
        // cooperative load of W[kc .. kc+64) x 128 into padded LDS
        #pragma unroll
        for (int i = 0; i < 8; i++) {
            int idx = t + i * 256;        // 0..2047 float4 slots
            int r   = idx >> 5;           // 0..63
            int cq  = idx & 31;           // 0..31
            v4f wv = *(const v4f*)(W + (long long)(kc + r) * D + cq * 4);
            *(v4f*)(sW + r * LDW + cq * 4) = wv;
        }
        __syncthreads();

        if (active) {
            const float* src = (kc < 128) ? X : AGG;
            const bool scale = (kc >= 128);
            // lane's A row and starting K within this source
            const long long rowoff = (long long)(m0 + nl) * D + (kc & 127) + 2 * half;
            #pragma unroll 4
            for (int k = 0; k < 64; k += 4) {
                v2f a = *(const v2f*)(src + rowoff + k);
                if (scale) { a.x *= invd; a.y *= invd; }
                #pragma unroll
                for (int nt = 0; nt < 8; nt++) {
                    v2f b;
                    b.x = sW[(k + 2 * half + 0) * LDW + nt * 16 + nl];
                    b.y = sW[(k + 2 * half + 1) * LDW + nt * 16 + nl];
                    acc[nt] = __builtin_amdgcn_wmma_f32_16x16x4_f32(
                        /*neg_a=*/false, a, /*neg_b=*/false, b,
                        /*c_mod=*/(short)0, acc[nt],
                        /*reuse_a=*/false, /*reuse_b=*/false);
                }
            }
        }
        __syncthreads();
    }

    if (active) {
        // C/D layout: VGPR r -> row m0 + r + 8*half, col nt*16 + nl
        #pragma unroll
        for (int nt = 0; nt < 8; nt++) {
            const int colIdx = nt * 16 + nl;
            const float bv = bias[colIdx];
            #pragma unroll
            for (int r = 0; r < 8; r++) {
                float v = acc[nt][r] + bv;
                if (relu) v = fmaxf(v, 0.0f);
                OUT[(long long)(m0 + r + 8 * half) * D + colIdx] = v;
            }
        }
    }
}

// ---------------- host-side orchestration ----------------
extern "C" void kernel_launch(void* const* d_in, const int* in_sizes, int n_in,
                              void* d_out, int out_size, void* d_ws, size_t ws_size,
                              hipStream_t stream) {
    const float* x   = (const float*)d_in[0];
    const int*  edge = (const int*)d_in[1];
    const float* W1  = (const float*)d_in[2];
    const float* b1  = (const float*)d_in[3];
    const float* W2  = (const float*)d_in[4];
    const float* b2  = (const float*)d_in[5];
    float* out = (float*)d_out;

    const int n  = in_sizes[0] / D;       // 100000 nodes
    const int nE = in_sizes[1] / 2;       // 1M edges
    const int* rowIdx = edge;             // edge_index[0] = destination
    const int* colIdx = edge + nE;        // edge_index[1] = source

    // workspace layout
    float* agg = (float*)d_ws;                         // n*D floats
    float* h   = agg + (long long)n * D;               // n*D floats
    float* inv = h + (long long)n * D;                 // n floats
    int*   deg = (int*)(inv + n);                      // n ints

    const long long featN4 = ((long long)n * D) / 4;

    // degree + inv_deg (once; reused by both layers)
    zero_v4<<<1024, 256, 0, stream>>>((float*)deg, (n + 3) / 4);
    degree_kernel<<<(nE + 255) / 256, 256, 0, stream>>>(rowIdx, deg, nE);
    invdeg_kernel<<<(n + 255) / 256, 256, 0, stream>>>(deg, inv, n);

    const int gemmBlocks = (n + 127) / 128;

    // ---- layer 1: h = relu([x | mean_agg(x)] @ W1 + b1) ----
    zero_v4<<<8192, 256, 0, stream>>>(agg, featN4);
    agg_kernel<<<2048, 256, 0, stream>>>(x, rowIdx, colIdx, agg, nE);
    sage_gemm<<<gemmBlocks, 256, 0, stream>>>(x, agg, inv, W1, b1, h, n, /*relu=*/1);

    // ---- layer 2: out = [h | mean_agg(h)] @ W2 + b2 ----
    zero_v4<<<8192, 256, 0, stream>>>(agg, featN4);
    agg_kernel<<<2048, 256, 0, stream>>>(h, rowIdx, colIdx, agg, nE);
    sage_gemm<<<gemmBlocks, 256, 0, stream>>>(h, agg, inv, W2, b2, out, n, /*relu=*/0);
}